// ParabolicPool1DFast_15161234555499
// MI455X (gfx1250) — compile-verified
//
#include <hip/hip_runtime.h>
#include <cstdint>

// Problem constants (from the reference): f (B=16, C=128, N=16384) f32,
// t (128,) f32, KS=25, STRIDE=2 -> out (16, 128, 8192) f32.
#define KS        25
#define HALF      12          // KS/2
#define N_IN      16384
#define N_OUT     8192
#define TILE_OUT  1024        // outputs per block
#define TILE_IN   (TILE_OUT * 2)          // 2048 inputs covered
#define SM_N      (TILE_IN + KS - 1)      // 2072 floats staged in LDS
#define NV4       ((SM_N + 3) / 4)        // 518 vec4 chunks
#define NTHREADS  256
#define OPT       4           // outputs per thread
#define WINW      ((OPT - 1) * 2 + KS)    // 31-float register window

__global__ __launch_bounds__(NTHREADS)
void parabolic_pool1d_maxplus_kernel(const float* __restrict__ f,
                                     const float* __restrict__ t,
                                     float* __restrict__ out,
                                     int C) {
    __shared__ float sm[SM_N + 8];

    const int tid  = threadIdx.x;
    const int tile = blockIdx.x;               // 0..7
    const int row  = blockIdx.y;               // b*C + c, 0..2047
    const int c    = row % C;

    const float* frow = f + (size_t)row * N_IN;
    const int in_start = tile * TILE_IN - HALF;   // first global input idx of sm[0]

    // -------- Stage tile + halo into LDS with CDNA5 async global->LDS copies.
    // GVS addressing: saddr = row base (SGPR pair), vaddr = per-lane byte offset.
    // Addresses are clamped into [0, N-4] (4-float aligned) so edge-tile halo
    // loads never leave the row; clamped-garbage slots are fixed up below.
    const uint64_t base   = (uint64_t)(uintptr_t)frow;
    const uint32_t smbase = (uint32_t)(uintptr_t)&sm[0];   // LDS byte offset
#pragma unroll
    for (int it = 0; it < 3; ++it) {                       // 3*256 >= 518 vec4s
        int vec = it * NTHREADS + tid;
        if (vec > NV4 - 1) vec = NV4 - 1;                  // redundant dup: benign
        int e = in_start + vec * 4;                        // element idx (mult of 4)
        if (e < 0) e = 0;
        if (e > N_IN - 4) e = N_IN - 4;
        uint32_t goff = (uint32_t)(e * 4);                 // byte offset in row
        uint32_t loff = smbase + (uint32_t)(vec * 16);     // LDS byte address
        asm volatile("global_load_async_to_lds_b128 %0, %1, %2"
                     :: "v"(loff), "v"(goff), "s"(base) : "memory");
    }
    asm volatile("s_wait_asynccnt 0" ::: "memory");        // own ASYNCcnt drained
    __syncthreads();                                       // all waves' data in LDS

    // -------- -inf padding fixup for out-of-row halo (only edge tiles hit this).
    if (tid < HALF) {
        if (in_start + tid < 0)            sm[tid]            = -INFINITY;
        int j2 = SM_N - 1 - tid;
        if (in_start + j2 >= N_IN)         sm[j2]             = -INFINITY;
    }
    __syncthreads();

    // -------- Per-channel parabolic taps: h[d] = -(d-12)^2 / (4 t[c]).
    const float inv = -0.25f / t[c];
    float hh[KS];
#pragma unroll
    for (int d = 0; d < KS; ++d) {
        const int z = d - HALF;
        hh[d] = (float)(z * z) * inv;
    }

    // -------- Max-plus reduction: 4 stride-2 outputs from a 31-float window.
    const int wbase = tid * (OPT * 2);                     // 8*tid words into sm
    float w[WINW];
#pragma unroll
    for (int i = 0; i < WINW; ++i) w[i] = sm[wbase + i];

    float acc[OPT];
#pragma unroll
    for (int k = 0; k < OPT; ++k) acc[k] = -INFINITY;
#pragma unroll
    for (int d = 0; d < KS; ++d) {
#pragma unroll
        for (int k = 0; k < OPT; ++k)
            acc[k] = fmaxf(acc[k], w[k * 2 + d] + hh[d]);
    }

    // -------- Coalesced 128-bit store.
    float4 r = make_float4(acc[0], acc[1], acc[2], acc[3]);
    *(float4*)(out + (size_t)row * N_OUT + tile * TILE_OUT + tid * OPT) = r;
}

extern "C" void kernel_launch(void* const* d_in, const int* in_sizes, int n_in,
                              void* d_out, int out_size, void* d_ws, size_t ws_size,
                              hipStream_t stream) {
    const float* f = (const float*)d_in[0];
    const float* t = (const float*)d_in[1];
    float* out     = (float*)d_out;

    const int C    = in_sizes[1];              // 128
    const int rows = in_sizes[0] / N_IN;       // B*C = 2048
    dim3 grid(N_IN / TILE_IN, rows);           // (8, 2048)
    parabolic_pool1d_maxplus_kernel<<<grid, NTHREADS, 0, stream>>>(f, t, out, C);
}